// CharAttention_42597485642385
// MI455X (gfx1250) — compile-verified
//
#include <hip/hip_runtime.h>

// Char-level causal attention, B=512, W=128, c=24, C=32, H=2, D=16.
// One wave32 per (b,w) pair; K/V GEMM via V_WMMA_F32_16X16X4_F32 (native fp32),
// scores/softmax/AV/proj via wave shuffles; output only at gather index.
// Next-pair x-tile pulled in with global_prefetch while the VALU tail runs.

#define BDIM   512
#define WDIM   128
#define CBLK   24
#define CEMB   32
#define NPAIR  (BDIM * WDIM)          // 65536

#define NBLOCKS          2048
#define WAVES_PER_BLOCK  8
#define NTHREADS         (WAVES_PER_BLOCK * 32)
#define PAIRS_PER_WAVE   (NPAIR / (NBLOCKS * WAVES_PER_BLOCK))  // 4

#define KV_STRIDE 65                  // padded LDS row stride (floats), odd -> conflict-free

typedef __attribute__((ext_vector_type(2))) float v2f;
typedef __attribute__((ext_vector_type(8))) float v8f;

#define NEG_BIG (-3.0e38f)

__global__ __launch_bounds__(NTHREADS)
void CharAttention_gfx1250_kernel(const float* __restrict__ x,
                                  const int*   __restrict__ x_end_idx,
                                  const float* __restrict__ w_attn,
                                  const float* __restrict__ w_proj,
                                  float*       __restrict__ out)
{
    __shared__ float wA[CEMB * 96];                       // 12 KB  (q-proj reads)
    __shared__ float wP[CEMB * CEMB];                     //  4 KB  (out-proj reads)
    __shared__ float kvs[WAVES_PER_BLOCK][CBLK * KV_STRIDE];  // per-wave K|V buffer

    const int tid = threadIdx.x;

    for (int i = tid; i < CEMB * 96; i += NTHREADS)  wA[i] = w_attn[i];
    for (int i = tid; i < CEMB * CEMB; i += NTHREADS) wP[i] = w_proj[i];
    __syncthreads();

    const int wid   = tid >> 5;
    const int lane  = tid & 31;
    const int lhalf = lane & 15;
    const int khi   = (lane >> 4) << 1;      // 0 for lanes 0-15, 2 for lanes 16-31

    float* kv = kvs[wid];

    // -------- WMMA B operands: w_attn cols 32..95 (K then V, per head) --------
    // B 4x16 tile layout: lane<16 -> N=lane, K=0,1 in {.x,.y}; lane>=16 -> N=lane-16, K=2,3.
    // breg[t][s]: N-tile t in {K.h0, K.h1, V.h0, V.h1}, k-step s covers k = 4s..4s+3.
    v2f breg[4][8];
    #pragma unroll
    for (int t = 0; t < 4; ++t) {
        const int col = 32 + 16 * t + lhalf;
        #pragma unroll
        for (int s = 0; s < 8; ++s) {
            const int k0 = 4 * s + khi;
            v2f b; b.x = w_attn[k0 * 96 + col]; b.y = w_attn[(k0 + 1) * 96 + col];
            breg[t][s] = b;
        }
    }

    const int gwave = blockIdx.x * WAVES_PER_BLOCK + wid;

    for (int p = 0; p < PAIRS_PER_WAVE; ++p) {
        const int    pair = gwave * PAIRS_PER_WAVE + p;
        const float* xp   = x + (size_t)pair * (CBLK * CEMB);
        const int    idx  = x_end_idx[pair];               // uniform across the wave
        const int    mtiles = (idx < 16) ? 1 : 2;          // skip rows 16..23 when unused

        // Prefetch next pair's 3 KB x-tile: 24 lanes each touch one 128 B row.
        if (p + 1 < PAIRS_PER_WAVE && lane < CBLK) {
            __builtin_prefetch(xp + (CBLK + lane) * CEMB, 0, 3);
        }

        // ---------------- K,V = X @ w_attn[:,32:96] via fp32 WMMA ----------------
        for (int m = 0; m < mtiles; ++m) {
            const int  row    = m * 16 + lhalf;
            const bool rvalid = (row < CBLK);

            // A 16x4 tile layout: lane<16 -> M=lane, K=0,1; lane>=16 -> M=lane-16, K=2,3.
            v2f areg[8];
            #pragma unroll
            for (int s = 0; s < 8; ++s) {
                v2f a;
                if (rvalid) {
                    const float* ap = xp + row * CEMB + 4 * s + khi;
                    a.x = ap[0]; a.y = ap[1];
                } else {
                    a.x = 0.0f; a.y = 0.0f;                // pad rows 24..31 with zeros
                }
                areg[s] = a;
            }

            v8f acc0 = {0.f,0.f,0.f,0.f,0.f,0.f,0.f,0.f};
            v8f acc1 = acc0, acc2 = acc0, acc3 = acc0;
            #pragma unroll
            for (int s = 0; s < 8; ++s) {
                acc0 = __builtin_amdgcn_wmma_f32_16x16x4_f32(false, areg[s], false, breg[0][s],
                                                             (short)0, acc0, false, false);
                acc1 = __builtin_amdgcn_wmma_f32_16x16x4_f32(false, areg[s], false, breg[1][s],
                                                             (short)0, acc1, false, false);
                acc2 = __builtin_amdgcn_wmma_f32_16x16x4_f32(false, areg[s], false, breg[2][s],
                                                             (short)0, acc2, false, false);
                acc3 = __builtin_amdgcn_wmma_f32_16x16x4_f32(false, areg[s], false, breg[3][s],
                                                             (short)0, acc3, false, false);
            }

            // D 16x16 layout: VGPR i holds (M = mbase+i, N = lane%16), mbase = 0 / 8.
            const int rbase = m * 16 + ((lane < 16) ? 0 : 8);
            #pragma unroll
            for (int i = 0; i < 8; ++i) {
                const int r = rbase + i;
                if (r < CBLK) {
                    float* krow = kv + r * KV_STRIDE + lhalf;
                    krow[0]  = acc0[i];   // K head0 -> cols  0..15
                    krow[16] = acc1[i];   // K head1 -> cols 16..31
                    krow[32] = acc2[i];   // V head0 -> cols 32..47
                    krow[48] = acc3[i];   // V head1 -> cols 48..63
                }
            }
        }

        // ---------------- single-row q / scores / softmax / AV / proj ----------------
        const float xv = xp[idx * CEMB + lane];            // residual row, lane-private

        // q[lane] = sum_k x[idx][k] * Wq[k][lane]   (Wq = w_attn cols 0..31)
        float q = 0.0f;
        #pragma unroll
        for (int k = 0; k < CEMB; ++k) {
            q += __shfl(xv, k, 32) * wA[k * 96 + lane];
        }

        // lane j = key index; two heads in parallel registers
        const int  jr    = (lane < CBLK) ? lane : 0;
        const bool valid = (lane <= idx);                  // idx < 24, implies lane < CBLK
        float s0 = 0.0f, s1 = 0.0f;
        #pragma unroll
        for (int d = 0; d < 16; ++d) {
            const float q0 = __shfl(q, d, 32);
            const float q1 = __shfl(q, 16 + d, 32);
            s0 += q0 * kv[jr * KV_STRIDE + d];
            s1 += q1 * kv[jr * KV_STRIDE + 16 + d];
        }
        s0 = valid ? s0 * 0.25f : NEG_BIG;                 // scale = 1/sqrt(16)
        s1 = valid ? s1 * 0.25f : NEG_BIG;

        float m0 = s0, m1 = s1;
        #pragma unroll
        for (int off = 16; off > 0; off >>= 1) {
            m0 = fmaxf(m0, __shfl_xor(m0, off, 32));
            m1 = fmaxf(m1, __shfl_xor(m1, off, 32));
        }
        const float e0 = __expf(s0 - m0);                  // masked lanes -> 0
        const float e1 = __expf(s1 - m1);
        float sum0 = e0, sum1 = e1;
        #pragma unroll
        for (int off = 16; off > 0; off >>= 1) {
            sum0 += __shfl_xor(sum0, off, 32);
            sum1 += __shfl_xor(sum1, off, 32);
        }
        const float att0 = e0 / sum0;
        const float att1 = e1 / sum1;

        // o[lane] = sum_{j<=idx} att_h[j] * V[j][lane]   (lane = h*16 + d, V at cols 32+)
        float o = 0.0f;
        for (int j = 0; j <= idx; ++j) {
            const float a0 = __shfl(att0, j, 32);
            const float a1 = __shfl(att1, j, 32);
            const float av = (lane < 16) ? a0 : a1;
            o += av * kv[j * KV_STRIDE + 32 + lane];
        }

        // y[lane] = sum_k o[k] * w_proj[k][lane]; residual; store gathered row
        float y = 0.0f;
        #pragma unroll
        for (int k = 0; k < CEMB; ++k) {
            y += __shfl(o, k, 32) * wP[k * CEMB + lane];
        }
        out[(size_t)pair * CEMB + lane] = xv + y;
    }
}

extern "C" void kernel_launch(void* const* d_in, const int* in_sizes, int n_in,
                              void* d_out, int out_size, void* d_ws, size_t ws_size,
                              hipStream_t stream) {
    (void)in_sizes; (void)n_in; (void)out_size; (void)d_ws; (void)ws_size;
    const float* x      = (const float*)d_in[0];
    const int*   endidx = (const int*)  d_in[1];
    const float* w_attn = (const float*)d_in[2];
    const float* w_proj = (const float*)d_in[3];
    float*       out    = (float*)d_out;

    hipLaunchKernelGGL(CharAttention_gfx1250_kernel,
                       dim3(NBLOCKS), dim3(NTHREADS), 0, stream,
                       x, endidx, w_attn, w_proj, out);
}